// DescartesExtension_4750233829894
// MI455X (gfx1250) — compile-verified
//
#include <hip/hip_runtime.h>

typedef __attribute__((ext_vector_type(2))) float v2f;
typedef __attribute__((ext_vector_type(8))) float v8f;
typedef int v4i __attribute__((vector_size(16)));   // matches builtin's V4i param

#define DIM      512
#define NT       32          // 16-wide tiles per dimension
#define NTILES   528         // NT*(NT+1)/2 upper-triangular tiles
#define NPAIRS   131328      // DIM*(DIM+1)/2
#define BLOCK    256         // 8 waves (wave32)

#if defined(__gfx1250__)
#  if __has_builtin(__builtin_amdgcn_global_load_async_to_lds_b128) && \
      __has_builtin(__builtin_amdgcn_s_wait_asynccnt)
#    define USE_ASYNC_LDS 1
#  endif
#endif
#ifndef USE_ASYNC_LDS
#  define USE_ASYNC_LDS 0
#endif

__global__ __launch_bounds__(BLOCK)
void triu_outer_wmma_kernel(const float* __restrict__ x, float* __restrict__ out) {
  __shared__ float row[DIM];

  const int b   = blockIdx.x;
  const int tid = threadIdx.x;

  // Stage this batch row (2 KB) into LDS.
#if USE_ASYNC_LDS
  // CDNA5 async copy: 128 x b128 transfers, tracked on ASYNCcnt.
  if (tid < DIM / 4) {
    __builtin_amdgcn_global_load_async_to_lds_b128(
        (__attribute__((address_space(1))) v4i*)(x + (size_t)b * DIM + tid * 4),
        (__attribute__((address_space(3))) v4i*)(&row[tid * 4]),
        /*offset=*/0, /*cpol=*/0);
  }
  __builtin_amdgcn_s_wait_asynccnt(0);
  __syncthreads();
#else
  {
    const float2* src = (const float2*)(x + (size_t)b * DIM);
    float2 v = src[tid];
    row[2 * tid]     = v.x;
    row[2 * tid + 1] = v.y;
  }
  __syncthreads();
#endif

  // Wave-uniform wave id -> tile index lives in SGPRs (scalar decode, saddr stores).
  const int wave = __builtin_amdgcn_readfirstlane(tid >> 5);
  const int lane = tid & 31;
  float* outb = out + (size_t)b * NPAIRS;

  for (int t = wave; t < NTILES; t += (BLOCK / 32)) {
    // Decode linear upper-tri tile index t -> (ti, tj), ti <= tj.
    int u = (NTILES - 1) - t;
    int r = (int)((sqrtf((float)(8 * u + 1)) - 1.0f) * 0.5f);
    while ((r + 1) * (r + 2) / 2 <= u) ++r;   // guard fp sqrt rounding
    while (r * (r + 1) / 2 > u) --r;
    const int ti = (NT - 1) - r;
    const int tj = (NT - 1) - (u - r * (r + 1) / 2);

    const int m0 = ti * 16;
    const int n0 = tj * 16;

    // Branchless operand build: all 32 lanes read LDS (lanes 16-31 broadcast),
    // then v_cndmask zeroes the K=1..3 slices. Rank-1 product via
    // V_WMMA_F32_16X16X4_F32 with vectors in the K=0 slice.
    const float av = row[m0 + (lane & 15)];
    const float bv = row[n0 + (lane & 15)];
    const bool lo = (lane < 16);
    v2f A, B;
    A.x = lo ? av : 0.0f;  A.y = 0.0f;
    B.x = lo ? bv : 0.0f;  B.y = 0.0f;
    v8f C = {};
    v8f Dt = __builtin_amdgcn_wmma_f32_16x16x4_f32(
        /*neg_a=*/false, A, /*neg_b=*/false, B,
        /*c_mod=*/(short)0, C, /*reuse_a=*/false, /*reuse_b=*/false);

    // D layout: VGPR k holds row M = m0 + k (+8 for lanes 16-31), col N = n0 + lane%16.
    const int col   = n0 + (lane & 15);
    const int rbase = m0 + ((lane >> 4) << 3);

    // Packed triangular offset, 32-bit math, strength-reduced across k:
    // off(ri) = ri*DIM - ri*(ri-1)/2 + (col - ri);  off(ri+1) = off(ri) + DIM-1-ri.
    int off = rbase * DIM - ((rbase * (rbase - 1)) >> 1) + (col - rbase);

    if (ti == tj) {
      #pragma unroll
      for (int k = 0; k < 8; ++k) {
        const int ri = rbase + k;
        if (col >= ri) __builtin_nontemporal_store(Dt[k], outb + off);
        off += DIM - 1 - ri;
      }
    } else {
      #pragma unroll
      for (int k = 0; k < 8; ++k) {
        __builtin_nontemporal_store(Dt[k], outb + off);
        off += DIM - 1 - (rbase + k);
      }
    }
  }
}

extern "C" void kernel_launch(void* const* d_in, const int* in_sizes, int n_in,
                              void* d_out, int out_size, void* d_ws, size_t ws_size,
                              hipStream_t stream) {
  (void)n_in; (void)out_size; (void)d_ws; (void)ws_size;
  const float* x = (const float*)d_in[0];
  float* out = (float*)d_out;
  const int B = in_sizes[0] / DIM;   // 1024 for the reference shapes
  triu_outer_wmma_kernel<<<dim3(B), dim3(BLOCK), 0, stream>>>(x, out);
}